// SimGNN_84482006712593
// MI455X (gfx1250) — compile-verified
//
#include <hip/hip_runtime.h>
#include <hip/hip_bf16.h>
#include <math.h>

typedef float v2f __attribute__((ext_vector_type(2)));
typedef float v8f __attribute__((ext_vector_type(8)));

#define NPG 512
#define HD 64
#define BINS 16

// ---- order-preserving float<->uint encoding for atomicMax on floats ----
__device__ __forceinline__ unsigned enc_f32(float x) {
  unsigned u = __float_as_uint(x);
  return (u & 0x80000000u) ? ~u : (u | 0x80000000u);
}
__device__ __forceinline__ float dec_f32(unsigned e) {
  unsigned u = (e & 0x80000000u) ? (e & 0x7FFFFFFFu) : ~e;
  return __uint_as_float(u);
}

// ---------------- simple float4 copy (init acc = x) ----------------
__global__ void copy4_kernel(float4* __restrict__ dst, const float4* __restrict__ src, long n4) {
  long i = (long)blockIdx.x * blockDim.x + threadIdx.x;
  if (i < n4) dst[i] = src[i];
}

// ---------------- edge aggregation: acc[dst] += x[src] ----------------
// Node tables fit in 192MB L2 -> gathers and atomics stay on-chip.
template<int F4>
__global__ void edge_agg_kernel(const float* __restrict__ x, const int* __restrict__ edges,
                                float* __restrict__ acc, int nE) {
  long idx = (long)blockIdx.x * blockDim.x + threadIdx.x;
  long tot = (long)nE * F4;
  if (idx >= tot) return;
  int e = (int)(idx / F4);
  int c = (int)(idx % F4);
  int s = edges[e];         // e[0][:]
  int d = edges[nE + e];    // e[1][:]
  const float4 v = ((const float4*)(x + (long)s * (F4 * 4)))[c];
  float* p = acc + (long)d * (F4 * 4) + (long)c * 4;
  atomicAdd(p + 0, v.x); atomicAdd(p + 1, v.y);
  atomicAdd(p + 2, v.z); atomicAdd(p + 3, v.w);
}

// ---------------- fused 2-layer GIN MLP via V_WMMA_F32_16X16X4_F32 ----------------
// out = relu( relu(h@W1+b1) @ W2 + b2 ), h:[N,FIN], W1:[FIN,64], W2:[64,64]
// 128 threads = 4 waves; block handles 16 rows; wave w -> output cols [16w,16w+16).
template<int FIN>
__global__ __launch_bounds__(128) void gin_mlp_kernel(
    const float* __restrict__ hin,
    const float* __restrict__ W1, const float* __restrict__ b1,
    const float* __restrict__ W2, const float* __restrict__ b2,
    float* __restrict__ out) {
  __shared__ float sW1[FIN * HD];
  __shared__ float sW2[HD * HD];
  __shared__ float sT[16 * HD];
  int tid = threadIdx.x;
  for (int i = tid; i < FIN * HD; i += 128) sW1[i] = W1[i];
  for (int i = tid; i < HD * HD; i += 128) sW2[i] = W2[i];
  __syncthreads();

  int wave = tid >> 5, lane = tid & 31;
  int nloc = lane & 15, khalf = lane >> 4;   // A: lane=M, half selects K pair
  int n = wave * 16 + nloc;                  // output column
  long row0 = (long)blockIdx.x * 16;

  // ---- layer 1: t = relu(h@W1 + b1) ----
  float bias1 = b1[n];
  v8f c;
  #pragma unroll
  for (int r = 0; r < 8; ++r) c[r] = bias1;
  const float* arow = hin + (row0 + nloc) * FIN;
  #pragma unroll 4
  for (int k0 = 0; k0 < FIN; k0 += 4) {
    int ka = k0 + 2 * khalf;
    v2f a; a.x = arow[ka]; a.y = arow[ka + 1];
    v2f b; b.x = sW1[ka * HD + n]; b.y = sW1[(ka + 1) * HD + n];
    c = __builtin_amdgcn_wmma_f32_16x16x4_f32(false, a, false, b, (short)0, c, false, false);
  }
  #pragma unroll
  for (int r = 0; r < 8; ++r) {
    int m = r + 8 * khalf;                   // C: vgpr r = rows r / r+8
    sT[m * HD + n] = fmaxf(c[r], 0.0f);
  }
  __syncthreads();

  // ---- layer 2: out = relu(t@W2 + b2) ----
  float bias2 = b2[n];
  v8f c2;
  #pragma unroll
  for (int r = 0; r < 8; ++r) c2[r] = bias2;
  const float* trow = sT + nloc * HD;
  #pragma unroll 4
  for (int k0 = 0; k0 < HD; k0 += 4) {
    int ka = k0 + 2 * khalf;
    v2f a; a.x = trow[ka]; a.y = trow[ka + 1];
    v2f b; b.x = sW2[ka * HD + n]; b.y = sW2[(ka + 1) * HD + n];
    c2 = __builtin_amdgcn_wmma_f32_16x16x4_f32(false, a, false, b, (short)0, c2, false, false);
  }
  #pragma unroll
  for (int r = 0; r < 8; ++r) {
    int m = r + 8 * khalf;
    out[(row0 + m) * HD + n] = fmaxf(c2[r], 0.0f);
  }
}

// ---------------- row L2-normalize (one wave per row, dim 64) ----------------
__global__ void normalize_kernel(const float* __restrict__ h, float* __restrict__ o, int nRows) {
  int gw = (int)(((long)blockIdx.x * blockDim.x + threadIdx.x) >> 5);
  int lane = threadIdx.x & 31;
  if (gw >= nRows) return;
  float2 v = ((const float2*)(h + (long)gw * HD))[lane];
  float ss = v.x * v.x + v.y * v.y;
  #pragma unroll
  for (int off = 16; off; off >>= 1) ss += __shfl_xor(ss, off, 32);
  float inv = 1.0f / fmaxf(sqrtf(ss), 1e-12f);
  float2 r; r.x = v.x * inv; r.y = v.y * inv;
  ((float2*)(o + (long)gw * HD))[lane] = r;
}

// ---------------- global mean pool ----------------
__global__ void pool_kernel(const float* __restrict__ h, float* __restrict__ g, int nOut) {
  int idx = blockIdx.x * blockDim.x + threadIdx.x;
  if (idx >= nOut) return;
  int b = idx >> 6, f = idx & 63;
  const float* p = h + (long)b * NPG * HD + f;
  float s = 0.f;
  for (int i = 0; i < NPG; ++i) s += p[(long)i * HD];
  g[idx] = s * (1.0f / NPG);
}

// ---------------- zero per-graph accumulators (every launch) ----------------
__global__ void zero_accum_kernel(unsigned* hist, float* sum, float* sumsq, unsigned* maxenc, int Bg) {
  int i = blockIdx.x * blockDim.x + threadIdx.x;
  if (i < Bg * BINS) hist[i] = 0u;
  if (i < Bg) { sum[i] = 0.f; sumsq[i] = 0.f; maxenc[i] = 0u; }
}

// ---------------- fused sim-matrix + histogram + stats ----------------
// grid = Bg*32 blocks: (graph b, 16-row strip of E1). 256 threads = 8 waves.
// Each wave: hold A fragments for full K=64, sweep 4 column tiles of E2,
// 16 WMMAs per tile, bin/accumulate results without ever writing sim.
__global__ __launch_bounds__(256) void sim_hist_kernel(
    const float* __restrict__ E1n, const float* __restrict__ E2n,
    unsigned* __restrict__ hist, float* __restrict__ gsum,
    float* __restrict__ gsumsq, unsigned* __restrict__ gmax) {
  int b = blockIdx.x >> 5;
  int strip = blockIdx.x & 31;
  int tid = threadIdx.x, wave = tid >> 5, lane = tid & 31;
  int nloc = lane & 15, khalf = lane >> 4;

  __shared__ unsigned sHist[BINS];
  __shared__ float sSum[8], sSq[8], sMax[8];
  if (tid < BINS) sHist[tid] = 0u;
  __syncthreads();

  const float* arow = E1n + ((long)b * NPG + strip * 16 + nloc) * HD;
  v2f af[16];
  #pragma unroll
  for (int ks = 0; ks < 16; ++ks) {
    int ka = ks * 4 + 2 * khalf;
    af[ks].x = arow[ka]; af[ks].y = arow[ka + 1];
  }

  float lsum = 0.f, lsq = 0.f, lmax = -2.0f;
  for (int t = 0; t < 4; ++t) {
    int ct = wave * 4 + t;  // column tile 0..31
    const float* brow = E2n + ((long)b * NPG + ct * 16 + nloc) * HD;
    v8f c;
    #pragma unroll
    for (int r = 0; r < 8; ++r) c[r] = 0.0f;
    #pragma unroll
    for (int ks = 0; ks < 16; ++ks) {
      int ka = ks * 4 + 2 * khalf;
      v2f bb; bb.x = brow[ka]; bb.y = brow[ka + 1];
      c = __builtin_amdgcn_wmma_f32_16x16x4_f32(false, af[ks], false, bb, (short)0, c, false, false);
    }
    #pragma unroll
    for (int r = 0; r < 8; ++r) {
      float v = c[r];
      int bin = (int)floorf((v + 1.0f) * (BINS * 0.5f));
      bin = bin < 0 ? 0 : (bin > BINS - 1 ? BINS - 1 : bin);
      atomicAdd(&sHist[bin], 1u);
      lsum += v; lsq += v * v; lmax = fmaxf(lmax, v);
    }
  }
  #pragma unroll
  for (int off = 16; off; off >>= 1) {
    lsum += __shfl_xor(lsum, off, 32);
    lsq  += __shfl_xor(lsq,  off, 32);
    lmax = fmaxf(lmax, __shfl_xor(lmax, off, 32));
  }
  if (lane == 0) { sSum[wave] = lsum; sSq[wave] = lsq; sMax[wave] = lmax; }
  __syncthreads();
  if (tid == 0) {
    float ts = 0.f, tq = 0.f, tm = -2.f;
    for (int w = 0; w < 8; ++w) { ts += sSum[w]; tq += sSq[w]; tm = fmaxf(tm, sMax[w]); }
    atomicAdd(&gsum[b], ts);
    atomicAdd(&gsumsq[b], tq);
    atomicMax(&gmax[b], enc_f32(tm));
  }
  if (tid < BINS) atomicAdd(&hist[b * BINS + tid], sHist[tid]);
}

// ---------------- final feature assembly + 3-layer MLP (tiny) ----------------
__global__ void final_mlp_kernel(
    const float* __restrict__ G1, const float* __restrict__ G2,
    const unsigned* __restrict__ hist, const float* __restrict__ gsum,
    const float* __restrict__ gsumsq, const unsigned* __restrict__ gmax,
    const float* __restrict__ W1, const float* __restrict__ b1,
    const float* __restrict__ W2, const float* __restrict__ b2,
    const float* __restrict__ W3, const float* __restrict__ b3,
    float* __restrict__ out, int Bg) {
  int b = blockIdx.x * blockDim.x + threadIdx.x;
  if (b >= Bg) return;
  const int FINF = 2 * HD + BINS + 3;  // 147
  float f[2 * HD + BINS + 3];
  for (int i = 0; i < HD; ++i) { f[i] = G1[b * HD + i]; f[HD + i] = G2[b * HD + i]; }
  float hsum = 0.f;
  for (int j = 0; j < BINS; ++j) hsum += (float)hist[b * BINS + j];
  float hden = fmaxf(hsum, 1.0f);
  for (int j = 0; j < BINS; ++j) f[2 * HD + j] = (float)hist[b * BINS + j] / hden;
  float cnt = (float)NPG * (float)NPG;
  float mean = gsum[b] / cnt;
  float var = gsumsq[b] / cnt - mean * mean;
  f[2 * HD + BINS + 0] = mean;
  f[2 * HD + BINS + 1] = dec_f32(gmax[b]);
  f[2 * HD + BINS + 2] = sqrtf(fmaxf(var, 0.f));

  float t1[HD];
  for (int o = 0; o < HD; ++o) {
    float s = b1[o];
    for (int i = 0; i < FINF; ++i) s += f[i] * W1[i * HD + o];
    t1[o] = fmaxf(s, 0.f);
  }
  float t2[HD / 2];
  for (int o = 0; o < HD / 2; ++o) {
    float s = b2[o];
    for (int i = 0; i < HD; ++i) s += t1[i] * W2[i * (HD / 2) + o];
    t2[o] = fmaxf(s, 0.f);
  }
  float s = b3[0];
  for (int i = 0; i < HD / 2; ++i) s += t2[i] * W3[i];
  out[b] = s;
}

extern "C" void kernel_launch(void* const* d_in, const int* in_sizes, int n_in,
                              void* d_out, int out_size, void* d_ws, size_t ws_size,
                              hipStream_t stream) {
  const float* x1 = (const float*)d_in[0];
  const int*   e1 = (const int*)d_in[1];
  const float* x2 = (const float*)d_in[3];
  const int*   e2 = (const int*)d_in[4];
  const float* ew1 = (const float*)d_in[6];  const float* eb1 = (const float*)d_in[7];
  const float* ew2 = (const float*)d_in[8];  const float* eb2 = (const float*)d_in[9];
  const float* ew3 = (const float*)d_in[10]; const float* eb3 = (const float*)d_in[11];
  const float* ew4 = (const float*)d_in[12]; const float* eb4 = (const float*)d_in[13];
  const float* mw1 = (const float*)d_in[14]; const float* mb1 = (const float*)d_in[15];
  const float* mw2 = (const float*)d_in[16]; const float* mb2 = (const float*)d_in[17];
  const float* mw3 = (const float*)d_in[18]; const float* mb3 = (const float*)d_in[19];

  const int IN_ = 128;
  const int N  = in_sizes[0] / IN_;  // 65536
  const int E  = in_sizes[1] / 2;    // 1048576
  const int Bg = N / NPG;            // 128

  // ---- workspace layout (bump allocator, ~96MB) ----
  char* p = (char*)d_ws;
  float* P128 = (float*)p; p += (size_t)N * IN_ * sizeof(float);
  float* PA   = (float*)p; p += (size_t)N * HD * sizeof(float);
  float* PB   = (float*)p; p += (size_t)N * HD * sizeof(float);
  float* H1   = (float*)p; p += (size_t)N * HD * sizeof(float);
  float* H2   = (float*)p; p += (size_t)N * HD * sizeof(float);
  float* G1   = (float*)p; p += (size_t)Bg * HD * sizeof(float);
  float* G2   = (float*)p; p += (size_t)Bg * HD * sizeof(float);
  unsigned* hist = (unsigned*)p; p += (size_t)Bg * BINS * sizeof(unsigned);
  float* gsum = (float*)p; p += (size_t)Bg * sizeof(float);
  float* gsq  = (float*)p; p += (size_t)Bg * sizeof(float);
  unsigned* gmax = (unsigned*)p;

  auto run_side = [&](const float* x, const int* e, float* Hout) {
    long n4 = (long)N * IN_ / 4;
    copy4_kernel<<<(unsigned)((n4 + 255) / 256), 256, 0, stream>>>(
        (float4*)P128, (const float4*)x, n4);
    long tot1 = (long)E * (IN_ / 4);
    edge_agg_kernel<IN_ / 4><<<(unsigned)((tot1 + 255) / 256), 256, 0, stream>>>(x, e, P128, E);
    gin_mlp_kernel<IN_><<<N / 16, 128, 0, stream>>>(P128, ew1, eb1, ew2, eb2, PA);

    long n4b = (long)N * HD / 4;
    copy4_kernel<<<(unsigned)((n4b + 255) / 256), 256, 0, stream>>>(
        (float4*)PB, (const float4*)PA, n4b);
    long tot2 = (long)E * (HD / 4);
    edge_agg_kernel<HD / 4><<<(unsigned)((tot2 + 255) / 256), 256, 0, stream>>>(PA, e, PB, E);
    gin_mlp_kernel<HD><<<N / 16, 128, 0, stream>>>(PB, ew3, eb3, ew4, eb4, Hout);
  };
  run_side(x1, e1, H1);
  run_side(x2, e2, H2);

  pool_kernel<<<(Bg * HD + 255) / 256, 256, 0, stream>>>(H1, G1, Bg * HD);
  pool_kernel<<<(Bg * HD + 255) / 256, 256, 0, stream>>>(H2, G2, Bg * HD);

  // normalized embeddings overwrite PA/PB (no longer needed)
  normalize_kernel<<<(N * 32 + 255) / 256, 256, 0, stream>>>(H1, PA, N);
  normalize_kernel<<<(N * 32 + 255) / 256, 256, 0, stream>>>(H2, PB, N);

  zero_accum_kernel<<<(Bg * BINS + 255) / 256, 256, 0, stream>>>(hist, gsum, gsq, gmax, Bg);
  sim_hist_kernel<<<Bg * 32, 256, 0, stream>>>(PA, PB, hist, gsum, gsq, gmax);

  final_mlp_kernel<<<(Bg + 127) / 128, 128, 0, stream>>>(
      G1, G2, hist, gsum, gsq, gmax, mw1, mb1, mw2, mb2, mw3, mb3, (float*)d_out, Bg);
}